// TripletLoss_635655160601
// MI455X (gfx1250) — compile-verified
//
#include <hip/hip_runtime.h>
#include <hip/hip_bf16.h>

typedef float v2f __attribute__((ext_vector_type(2)));
typedef float v8f __attribute__((ext_vector_type(8)));

#define NROWS 8192
#define DIM   128
#define MARGIN_F 0.3f
#define BIGF  1e30f

// ---------------------------------------------------------------------------
// Kernel 0: per-row squared norms + zero the (sum, count) accumulators.
// ---------------------------------------------------------------------------
__global__ void triplet_prep(const float* __restrict__ f,
                             float* __restrict__ sq,
                             float* __restrict__ accum) {
    int row = blockIdx.x * blockDim.x + threadIdx.x;
    if (row == 0) { accum[0] = 0.0f; accum[1] = 0.0f; }
    if (row < NROWS) {
        const float4* p = (const float4*)(f + (size_t)row * DIM);
        float s = 0.0f;
#pragma unroll
        for (int i = 0; i < DIM / 4; ++i) {
            float4 v = p[i];
            s += v.x * v.x + v.y * v.y + v.z * v.z + v.w * v.w;
        }
        sq[row] = s;
    }
}

// Async global -> LDS copy of one 16-byte chunk (CDNA5 ASYNCcnt path).
// lds_off = byte offset within the wave's LDS allocation (low 32 bits of the
// generic shared-pointer address), gaddr = 64-bit global address.
__device__ __forceinline__ void async_b128_to_lds(unsigned lds_off,
                                                  const float4* gaddr) {
    asm volatile("global_load_async_to_lds_b128 %0, %1, off"
                 :: "v"(lds_off), "v"(gaddr)
                 : "memory");
}

__device__ __forceinline__ void wait_asynccnt_le2() {
    asm volatile("s_wait_asynccnt 2" ::: "memory");
}
__device__ __forceinline__ void wait_asynccnt_0() {
    asm volatile("s_wait_asynccnt 0" ::: "memory");
}

// ---------------------------------------------------------------------------
// Kernel 1: fused Gram (V_WMMA_F32_16X16X4_F32) + distance + batch-hard
// mining, with double-buffered async global->LDS streaming of the B tiles.
// Each wave owns a 16-anchor-row tile; block = 8 waves = 128 rows.
// ---------------------------------------------------------------------------
__global__ __launch_bounds__(256)
void triplet_main(const float* __restrict__ f,
                  const int*   __restrict__ labels,
                  const float* __restrict__ sq,
                  float*       __restrict__ accum) {
    __shared__ float ldsB[2][16 * DIM];       // 2 x 8 KB double buffer

    const int tid  = threadIdx.x;
    const int lane = tid & 31;                // wave32
    const int wave = tid >> 5;
    const int half = lane >> 4;               // 0: lanes 0-15, 1: lanes 16-31
    const int l16  = lane & 15;

    const int I    = blockIdx.x * 128 + wave * 16;   // anchor tile base row
    const int rowA = I + l16;

    // --- A fragments: lane l holds row l%16; {VGPR v, half h} -> K = 4s+2h+v
    v2f a[32];
#pragma unroll
    for (int s = 0; s < 32; ++s) {
        const float2 t = *(const float2*)(f + (size_t)rowA * DIM + 4 * s + 2 * half);
        a[s][0] = t.x; a[s][1] = t.y;
    }

    // --- per-lane metadata for the 8 C-rows it owns (row = v + 8*half)
    float sqI[8]; int labI[8];
#pragma unroll
    for (int v = 0; v < 8; ++v) {
        int r = I + v + 8 * half;
        sqI[v] = sq[r];
        labI[v] = labels[r];
    }

    float hp[8], hn[8];                       // hardest pos / hardest neg
#pragma unroll
    for (int v = 0; v < 8; ++v) { hp[v] = -BIGF; hn[v] = BIGF; }

    // LDS byte offsets of this thread's two 16B staging chunks in each buffer
    const unsigned ldsBase0 = (unsigned)(size_t)(&ldsB[0][0]) + (unsigned)tid * 16u;
    const unsigned ldsBase1 = (unsigned)(size_t)(&ldsB[1][0]) + (unsigned)tid * 16u;

    constexpr int NT = NROWS / 16;            // 512 column tiles

    // --- prologue: stream tile 0 into buffer 0 (2 async insts per wave-lane set)
    {
        const float4* g0 = (const float4*)f + tid;
        async_b128_to_lds(ldsBase0,          g0);
        async_b128_to_lds(ldsBase0 + 4096u,  g0 + 256);
    }

    for (int jt = 0; jt < NT; ++jt) {
        const int J = jt * 16;
        const float* curB = ldsB[jt & 1];

        // stream tile jt+1 into the other buffer while we compute tile jt
        if (jt + 1 < NT) {
            const float4* g0 = (const float4*)(f + (size_t)(J + 16) * DIM) + tid;
            const unsigned nxt = ((jt + 1) & 1) ? ldsBase1 : ldsBase0;
            async_b128_to_lds(nxt,         g0);
            async_b128_to_lds(nxt + 4096u, g0 + 256);
            wait_asynccnt_le2();              // batch jt (older, in-order) done
        } else {
            wait_asynccnt_0();
        }
        __syncthreads();                      // batch jt visible to all waves

        v8f c = {};
#pragma unroll
        for (int s = 0; s < 32; ++s) {
            const float2 t = *(const float2*)(curB + l16 * DIM + 4 * s + 2 * half);
            v2f b; b[0] = t.x; b[1] = t.y;
            c = __builtin_amdgcn_wmma_f32_16x16x4_f32(
                    false, a[s], false, b, (short)0, c, false, false);
        }
        __syncthreads();                      // seal reads of buf[jt&1]

        const float sqJ  = sq[J + l16];
        const int   labJ = labels[J + l16];
        const int   j    = J + l16;
#pragma unroll
        for (int v = 0; v < 8; ++v) {
            float d2   = sqI[v] + sqJ - 2.0f * c[v];
            float dist = sqrtf(fmaxf(d2, 1e-12f));
            int   i    = I + v + 8 * half;
            bool  same = (labI[v] == labJ);
            if (same && (i != j)) hp[v] = fmaxf(hp[v], dist);
            if (!same)            hn[v] = fminf(hn[v], dist);
        }
    }

    // --- reduce over the 16 lanes of each half (row set is per-half)
#pragma unroll
    for (int m = 1; m < 16; m <<= 1) {
#pragma unroll
        for (int v = 0; v < 8; ++v) {
            hp[v] = fmaxf(hp[v], __shfl_xor(hp[v], m, 32));
            hn[v] = fminf(hn[v], __shfl_xor(hn[v], m, 32));
        }
    }

    if (l16 == 0) {                           // lanes 0 and 16 emit 8 rows each
        float lsum = 0.0f, lcnt = 0.0f;
#pragma unroll
        for (int v = 0; v < 8; ++v) {
            bool  valid = (hp[v] > -1e29f) && (hn[v] < 1e29f);
            float loss  = fmaxf(hp[v] - hn[v] + MARGIN_F, 0.0f);
            if (valid) { lsum += loss; lcnt += 1.0f; }
        }
        atomicAdd(&accum[0], lsum);
        atomicAdd(&accum[1], lcnt);
    }
}

// ---------------------------------------------------------------------------
// Kernel 2: finalize scalar loss.
// ---------------------------------------------------------------------------
__global__ void triplet_finalize(const float* __restrict__ accum,
                                 float* __restrict__ out) {
    if (threadIdx.x == 0 && blockIdx.x == 0) {
        float s = accum[0], c = accum[1];
        out[0] = (c > 0.0f) ? (s / c) : 0.0f;
    }
}

// ---------------------------------------------------------------------------
extern "C" void kernel_launch(void* const* d_in, const int* in_sizes, int n_in,
                              void* d_out, int out_size, void* d_ws, size_t ws_size,
                              hipStream_t stream) {
    const float* f      = (const float*)d_in[0];
    const int*   labels = (const int*)d_in[1];   // labels in [0,512)
    float*       out    = (float*)d_out;

    float* sq    = (float*)d_ws;                 // NROWS floats
    float* accum = sq + NROWS;                   // 2 floats (sum, count)

    triplet_prep<<<NROWS / 256, 256, 0, stream>>>(f, sq, accum);
    triplet_main<<<NROWS / 128, 256, 0, stream>>>(f, labels, sq, accum);
    triplet_finalize<<<1, 32, 0, stream>>>(accum, out);
}